// DigitCapsLayer_30313879175433
// MI455X (gfx1250) — compile-verified
//
#include <hip/hip_runtime.h>
#include <math.h>

// Problem constants (from reference setup_inputs)
#define NCAP 10     // NC: digit capsules
#define BATCH 256   // B
#define ICAP 1152   // IC: input capsules
#define LDIM 8      // L: input capsule dim
#define OCAP 16     // OC: output capsule dim
#define NI (NCAP * ICAP)   // 11520

typedef __attribute__((ext_vector_type(2))) float v2f;
typedef __attribute__((ext_vector_type(8))) float v8f;

// ---------------------------------------------------------------------------
// Kernel 1: priors[n,i,b,o] = sum_l x[b,i,l] * w[n,i,l,o]
// Batched [256x8] @ [8x16] GEMM per (n,i) via V_WMMA_F32_16X16X4_F32.
// One wave per 16-row batch tile; 2 WMMAs chain K=0..3 and K=4..7.
// ---------------------------------------------------------------------------
__global__ __launch_bounds__(256)
void priors_wmma_kernel(const float* __restrict__ x,
                        const float* __restrict__ w,
                        float* __restrict__ priors)
{
    const int blk  = blockIdx.x;      // 0 .. NI*2-1
    const int hb   = blk & 1;         // which half of the batch (8 tiles each)
    const int ni   = blk >> 1;        // n*ICAP + i
    const int i    = ni % ICAP;
    const int wave = threadIdx.x >> 5;
    const int lane = threadIdx.x & 31;
    const int m    = lane & 15;              // A row / B,N column / D column
    const int kh   = (lane >> 4) << 1;       // lane-half K split: 0 or 2
    const int b0   = (hb * 8 + wave) * 16;   // batch tile base

    // A tile: x[b0+m, i, k] (per-lane 2 contiguous floats per WMMA)
    const float* xa = x + (size_t)(b0 + m) * (ICAP * LDIM) + (size_t)i * LDIM;
    v2f a0, a1;
    a0.x = xa[kh + 0]; a0.y = xa[kh + 1];
    a1.x = xa[kh + 4]; a1.y = xa[kh + 5];

    // B tile: w[n,i,k,o] is a contiguous 8x16 block; b[r] = B[K=r+kh][N=m]
    const float* wb = w + (size_t)ni * (LDIM * OCAP);
    v2f bv0, bv1;
    bv0.x = wb[(kh + 0) * OCAP + m]; bv0.y = wb[(kh + 1) * OCAP + m];
    bv1.x = wb[(kh + 4) * OCAP + m]; bv1.y = wb[(kh + 5) * OCAP + m];

    v8f acc = {};
    acc = __builtin_amdgcn_wmma_f32_16x16x4_f32(false, a0, false, bv0,
                                                (short)0, acc, false, false);
    acc = __builtin_amdgcn_wmma_f32_16x16x4_f32(false, a1, false, bv1,
                                                (short)0, acc, false, false);

    // D layout: d[r] = D[M = r + (lane<16?0:8)][N = m]
    float* p = priors + ((size_t)ni * BATCH + b0 + ((lane >> 4) << 3)) * OCAP + m;
#pragma unroll
    for (int r = 0; r < 8; ++r)
        p[(size_t)r * OCAP] = acc[r];
}

// ---------------------------------------------------------------------------
// Zero-init routing logits bl[n,i,b]
// ---------------------------------------------------------------------------
__global__ void zero_kernel(float* __restrict__ p, int n)
{
    int t = blockIdx.x * blockDim.x + threadIdx.x;
    if (t < n) p[t] = 0.0f;
}

// ---------------------------------------------------------------------------
// Kernel S1: c[n,i,b] = softmax over b (axis=1 of reference) of bl[n,i,b]
// One 256-thread block per (n,i); max-subtracted like jax.nn.softmax.
// ---------------------------------------------------------------------------
__global__ __launch_bounds__(256)
void softmax_kernel(const float* __restrict__ bl, float* __restrict__ c)
{
    const int ni = blockIdx.x;
    const int t  = threadIdx.x;   // batch index
    __shared__ float red[8];

    const float val = bl[(size_t)ni * BATCH + t];

    // block max
    float mx = val;
#pragma unroll
    for (int off = 16; off > 0; off >>= 1)
        mx = fmaxf(mx, __shfl_down(mx, off, 32));
    if ((t & 31) == 0) red[t >> 5] = mx;
    __syncthreads();
    mx = red[0];
#pragma unroll
    for (int j = 1; j < 8; ++j) mx = fmaxf(mx, red[j]);
    __syncthreads();

    // block sum of exp
    const float e = expf(val - mx);
    float sm = e;
#pragma unroll
    for (int off = 16; off > 0; off >>= 1)
        sm += __shfl_down(sm, off, 32);
    if ((t & 31) == 0) red[t >> 5] = sm;
    __syncthreads();
    sm = red[0];
#pragma unroll
    for (int j = 1; j < 8; ++j) sm += red[j];

    c[(size_t)ni * BATCH + t] = e / sm;
}

// ---------------------------------------------------------------------------
// Kernel S2: s[n,b,o] = sum_i c[n,i,b] * priors[n,i,b,o]
// Block per (n, 16-batch tile); thread = (b-in-tile, o); contiguous 1KB per i.
// ---------------------------------------------------------------------------
__global__ __launch_bounds__(256)
void s_kernel(const float* __restrict__ priors, const float* __restrict__ c,
              float* __restrict__ s)
{
    const int blk = blockIdx.x;        // n*16 + btile
    const int n   = blk >> 4;
    const int b   = ((blk & 15) << 4) + (threadIdx.x >> 4);
    const int o   = threadIdx.x & 15;

    const float* pp = priors + ((size_t)n * ICAP * BATCH + b) * OCAP + o;
    const float* cc = c + (size_t)n * ICAP * BATCH + b;

    float acc = 0.0f;
#pragma unroll 4
    for (int i = 0; i < ICAP; ++i)
        acc = fmaf(pp[(size_t)i * (BATCH * OCAP)], cc[(size_t)i * BATCH], acc);

    s[((size_t)n * BATCH + b) * OCAP + o] = acc;
}

// ---------------------------------------------------------------------------
// Kernel S3: global-Frobenius squash.  n2 = sum(s*s) over ALL elements,
// v = s * sqrt(n2)/(1+n2).  Single 1024-thread block; writes v + d_out.
// ---------------------------------------------------------------------------
__global__ __launch_bounds__(1024)
void squash_kernel(const float* __restrict__ s, float* __restrict__ v,
                   float* __restrict__ out)
{
    const int t = threadIdx.x;
    const int NELEM = NCAP * BATCH * OCAP;  // 40960
    __shared__ float red[32];

    float acc = 0.0f;
    for (int idx = t; idx < NELEM; idx += 1024) {
        float xv = s[idx];
        acc = fmaf(xv, xv, acc);
    }
#pragma unroll
    for (int off = 16; off > 0; off >>= 1)
        acc += __shfl_down(acc, off, 32);
    if ((t & 31) == 0) red[t >> 5] = acc;
    __syncthreads();
    if (t < 32) {
        float a = red[t];
#pragma unroll
        for (int off = 16; off > 0; off >>= 1)
            a += __shfl_down(a, off, 32);
        if (t == 0) red[0] = a;
    }
    __syncthreads();

    const float n2 = red[0];
    const float scale = sqrtf(n2) / (1.0f + n2);
    for (int idx = t; idx < NELEM; idx += 1024) {
        float val = s[idx] * scale;
        v[idx]   = val;
        out[idx] = val;   // final iteration's write is the answer
    }
}

// ---------------------------------------------------------------------------
// Kernel S4: bl[n,i,b] += sum_o priors[n,i,b,o] * v[n,b,o]
// Thread per (n,i,b): four float4 dots, 64B contiguous per lane.
// ---------------------------------------------------------------------------
__global__ __launch_bounds__(256)
void bupdate_kernel(const float* __restrict__ priors, const float* __restrict__ v,
                    float* __restrict__ bl)
{
    const size_t t  = (size_t)blockIdx.x * blockDim.x + threadIdx.x; // NCAP*ICAP*BATCH
    const int b     = (int)(t & (BATCH - 1));
    const size_t ni = t >> 8;
    const int n     = (int)(ni / ICAP);

    const float4* pp = (const float4*)(priors + ((size_t)ni * BATCH + b) * OCAP);
    const float4* vv = (const float4*)(v + ((size_t)n * BATCH + b) * OCAP);

    float acc = 0.0f;
#pragma unroll
    for (int q = 0; q < 4; ++q) {
        float4 p4 = pp[q], v4 = vv[q];
        acc = fmaf(p4.x, v4.x, acc);
        acc = fmaf(p4.y, v4.y, acc);
        acc = fmaf(p4.z, v4.z, acc);
        acc = fmaf(p4.w, v4.w, acc);
    }
    bl[t] += acc;
}

// ---------------------------------------------------------------------------
extern "C" void kernel_launch(void* const* d_in, const int* in_sizes, int n_in,
                              void* d_out, int out_size, void* d_ws, size_t ws_size,
                              hipStream_t stream)
{
    const float* x = (const float*)d_in[0];   // [256,1152,8]
    const float* w = (const float*)d_in[1];   // [10,1152,8,16]
    float* out = (float*)d_out;               // [10,256,16]

    // Workspace layout (floats)
    float* ws     = (float*)d_ws;
    float* priors = ws;                                            // NI*B*OC
    float* bl     = priors + (size_t)NI * BATCH * OCAP;            // NI*B
    float* c      = bl + (size_t)NI * BATCH;                       // NI*B
    float* s      = c + (size_t)NI * BATCH;                        // NC*B*OC
    float* v      = s + (size_t)NCAP * BATCH * OCAP;               // NC*B*OC

    // 1. priors via f32 WMMA
    priors_wmma_kernel<<<NI * 2, 256, 0, stream>>>(x, w, priors);

    // 2. zero routing logits
    zero_kernel<<<(NI * BATCH + 255) / 256, 256, 0, stream>>>(bl, NI * BATCH);

    // 3. three routing iterations (softmax of zeros == 1/256, so iter0 is exact)
    for (int it = 0; it < 3; ++it) {
        softmax_kernel<<<NI, 256, 0, stream>>>(bl, c);
        s_kernel<<<NCAP * 16, 256, 0, stream>>>(priors, c, s);
        squash_kernel<<<1, 1024, 0, stream>>>(s, v, out);
        if (it < 2)   // last iteration's b-update is dead code in the reference
            bupdate_kernel<<<(NI * BATCH) / 256, 256, 0, stream>>>(priors, v, bl);
    }
}